// GraphNet_67138928771130
// MI455X (gfx1250) — compile-verified
//
#include <hip/hip_runtime.h>

// ============================================================================
// GraphNet (message-passing GNN) for MI455X / gfx1250, fp32 WMMA path.
//
// All dense math runs through V_WMMA_F32_16X16X4_F32 (exact fp32, matches the
// fp32 reference). Each wave (wave32) owns a 16-row tile; feature dims are
// tiled N=16; K stepped by 4. Activations bounce through a wave-private LDS
// tile (stride TW+4 floats -> bank-conflict-free A reads & C writebacks).
//
// v2 changes vs round 1:
//  * removed K / column guards on B-fragment loads (padded-K columns of A are
//    zero in LDS and all "overshoot" W reads stay inside the params buffer,
//    so the unguarded math is exact for every stored column) -> kills the
//    divergent s_and_saveexec / per-element-load ladders seen in the asm.
//  * edge gather: per-row src/dst indices are preloaded once per lane and
//    broadcast with __shfl instead of re-reading global per element.
//
// d_in order (setup_inputs dict order):
//  0 params (flattened pytree, jax sorted-key order, 193332 f32)
//  1 branch_x [150000,6]   2 junction_x [40000,6]
//  3 inlet_feats [190000,7] 4 outlet_feats [190000,7]
//  5 e_b2b [400000,4] 6 e_j2j [100000,4] 7 e_b2j [100000,4] 8 e_j2b [100000,4]
//  9..16: b2b_src,b2b_dst,j2j_src,j2j_dst,b2j_src,b2j_dst,j2b_src,j2b_dst (i32)
// d_out: out_b [150000,2] then out_j [40000,2], fp32.
// ============================================================================

typedef __attribute__((ext_vector_type(2))) float v2f;
typedef __attribute__((ext_vector_type(8))) float v8f;

__device__ __forceinline__ float lrelu(float x) { return x > 0.f ? x : 0.01f * x; }

// C[16x(NT*16)] = X(LDS 16xKL) * W(global [KL x WROW]) + bias, f32 WMMA.
// A frag: lane (r,hi): A[r][k0+2*hi+{0,1}]. B frag: lane (r,hi): B[k0+2*hi+{0,1}][r].
// C frag: c[i] at lane (r,hi) = C[i+8*hi][r].
// NOTE: deliberately unguarded; callers guarantee A is zero-padded in LDS for
// K beyond the real feature count and that W/B overshoot stays in-bounds of
// the params allocation (garbage columns are never stored).
template<int KL, int NT, int WROW, int ST>
__device__ __forceinline__ void wmma_layer(const float* Xs,
                                           const float* __restrict__ W,
                                           const float* __restrict__ B,
                                           v8f* acc, int lane)
{
  const int r16 = lane & 15;
  const int hi  = lane >> 4;
#pragma unroll
  for (int nt = 0; nt < NT; ++nt) {
    const float bv = B[nt * 16 + r16];   // bias is per-column -> same in all 8
    v8f c;
#pragma unroll
    for (int i = 0; i < 8; ++i) c[i] = bv;
    acc[nt] = c;
  }
#pragma unroll
  for (int k0 = 0; k0 < KL; k0 += 4) {
    const int kr = k0 + 2 * hi;
    v2f a;
    a.x = Xs[r16 * ST + kr + 0];
    a.y = Xs[r16 * ST + kr + 1];
#pragma unroll
    for (int nt = 0; nt < NT; ++nt) {
      const int col = nt * 16 + r16;
      v2f b;
      b.x = W[(kr    ) * WROW + col];
      b.y = W[(kr + 1) * WROW + col];
      acc[nt] = __builtin_amdgcn_wmma_f32_16x16x4_f32(
          false, a, false, b, (short)0, acc[nt], false, false);
    }
  }
}

// GATHER: 0 = plain [rows,FI]; 1 = edge concat [e32 | x1[src]32 | x2[dst]32];
//         2 = node concat [p32 | pe32 | in8 | out8]
template<int FI, int FO, bool LN, bool RES, int GATHER>
__global__ __launch_bounds__(256) void mlp_wmma(
    const float* __restrict__ x0, const float* __restrict__ x1,
    const float* __restrict__ x2, const float* __restrict__ x3,
    const int*  __restrict__ sidx, const int* __restrict__ didx,
    const float* __restrict__ Wi, const float* __restrict__ bi,
    const float* __restrict__ Wh, const float* __restrict__ bh,
    const float* __restrict__ Wo, const float* __restrict__ bo,
    const float* __restrict__ lng, const float* __restrict__ lnb,
    const float* __restrict__ res, float* __restrict__ out, int nrows)
{
  constexpr int KP  = (FI + 3) & ~3;            // K padded to 4
  constexpr int TW  = (KP > 64) ? KP : 64;      // tile width (max of in/hidden)
  constexpr int ST  = TW + 4;                   // LDS row stride (bank-safe)
  constexpr int NTO = (FO + 15) / 16;
  __shared__ float tile[8][16 * ST];            // <= 51200 B
  const int wave = threadIdx.x >> 5;
  const int lane = threadIdx.x & 31;
  const int r16  = lane & 15;
  const int hi   = lane >> 4;
  float* Xs = tile[wave];
  const long row0 = (long)(blockIdx.x * 8 + wave) * 16;

  // edge mode: lanes 0..15 preload src idx of their row, lanes 16..31 dst idx
  int idxreg = 0;
  if constexpr (GATHER == 1) {
    const long rr = row0 + r16;
    if (rr < nrows) idxreg = (hi ? didx : sidx)[rr];
  }

  // ---- gather input tile [16][KP] into wave-private LDS (coalesced) ----
  for (int e = lane; e < 16 * KP; e += 32) {     // 16*KP is a multiple of 32
    const int r = e / KP, c = e % KP;
    int sIdx = 0, dIdx = 0;
    if constexpr (GATHER == 1) {                 // EXEC all-ones here
      sIdx = __shfl(idxreg, r, 32);
      dIdx = __shfl(idxreg, 16 + r, 32);
    }
    const long row = row0 + r;
    float v = 0.f;
    if (row < nrows) {
      if constexpr (GATHER == 0) {
        if (c < FI) v = x0[row * FI + c];
      } else if constexpr (GATHER == 1) {
        const int cc = c & 31;
        if      (c < 32) v = x0[row * 32 + cc];
        else if (c < 64) v = x1[(long)sIdx * 32 + cc];
        else             v = x2[(long)dIdx * 32 + cc];
      } else {
        if      (c < 32) v = x0[row * 32 + c];
        else if (c < 64) v = x1[row * 32 + (c - 32)];
        else if (c < 72) v = x2[row * 8  + (c - 64)];
        else             v = x3[row * 8  + (c - 72)];
      }
    }
    Xs[r * ST + c] = v;
  }
  // LDS ops from a single wave are processed in order; no barrier needed.

  // ---- layer 1: FI -> 64, leaky_relu ----
  v8f h[4];
  wmma_layer<KP, 4, 64, ST>(Xs, Wi, bi, h, lane);
#pragma unroll
  for (int nt = 0; nt < 4; ++nt)
#pragma unroll
    for (int i = 0; i < 8; ++i)
      Xs[(i + 8 * hi) * ST + nt * 16 + r16] = lrelu(h[nt][i]);

  // ---- layer 2: 64 -> 64, leaky_relu ----
  v8f h2[4];
  wmma_layer<64, 4, 64, ST>(Xs, Wh, bh, h2, lane);
#pragma unroll
  for (int nt = 0; nt < 4; ++nt)
#pragma unroll
    for (int i = 0; i < 8; ++i)
      Xs[(i + 8 * hi) * ST + nt * 16 + r16] = lrelu(h2[nt][i]);

  // ---- layer 3: 64 -> FO (garbage cols beyond FO are computed, never stored)
  v8f y[NTO];
  wmma_layer<64, NTO, FO, ST>(Xs, Wo, bo, y, lane);

  if constexpr (LN) {   // LayerNorm over FO==32 (row lives across 16 lanes x 2 tiles)
    const float g0  = lng[r16],      g1  = lng[16 + r16];
    const float be0 = lnb[r16],      be1 = lnb[16 + r16];
#pragma unroll
    for (int i = 0; i < 8; ++i) {
      float s = y[0][i] + y[1][i];
#pragma unroll
      for (int m = 1; m < 16; m <<= 1) s += __shfl_xor(s, m, 32);
      const float mu = s * (1.f / 32.f);
      const float d0 = y[0][i] - mu, d1 = y[1][i] - mu;
      float q = d0 * d0 + d1 * d1;
#pragma unroll
      for (int m = 1; m < 16; m <<= 1) q += __shfl_xor(q, m, 32);
      const float rs = rsqrtf(q * (1.f / 32.f) + 1e-5f);
      y[0][i] = d0 * rs * g0 + be0;
      y[1][i] = d1 * rs * g1 + be1;
    }
  }

  // ---- residual + store (coalesced per row) ----
#pragma unroll
  for (int nt = 0; nt < NTO; ++nt) {
    const int col = nt * 16 + r16;
    if (FO >= 16 || col < FO) {
#pragma unroll
      for (int i = 0; i < 8; ++i) {
        const long row = row0 + i + 8 * hi;
        if (row < nrows) {
          float v = y[nt][i];
          if constexpr (RES) v += res[row * FO + col];
          out[row * FO + col] = v;
        }
      }
    }
  }
}

// Tiny inlet/outlet encoder: 7 -> 16 -> 16 -> 8 + LN. Pure VALU (lat=16 too
// small for the matrix pipe to pay off).
__global__ __launch_bounds__(256) void enc_small_kernel(
    const float* __restrict__ x, const float* __restrict__ P,
    float* __restrict__ out, int nrows)
{
  const long row = (long)blockIdx.x * blockDim.x + threadIdx.x;
  if (row >= nrows) return;
  float in[7];
#pragma unroll
  for (int i = 0; i < 7; ++i) in[i] = x[row * 7 + i];
  const float* Wi = P + 288; const float* bi = P + 272;
  float h[16];
#pragma unroll
  for (int j = 0; j < 16; ++j) {
    float s = bi[j];
#pragma unroll
    for (int i = 0; i < 7; ++i) s = fmaf(in[i], Wi[i * 16 + j], s);
    h[j] = lrelu(s);
  }
  const float* Wh = P + 16; const float* bh = P + 0;
  float h2[16];
#pragma unroll
  for (int j = 0; j < 16; ++j) {
    float s = bh[j];
#pragma unroll
    for (int i = 0; i < 16; ++i) s = fmaf(h[i], Wh[i * 16 + j], s);
    h2[j] = lrelu(s);
  }
  const float* Wo = P + 424; const float* bo = P + 416;
  float y[8];
  float mu = 0.f;
#pragma unroll
  for (int j = 0; j < 8; ++j) {
    float s = bo[j];
#pragma unroll
    for (int i = 0; i < 16; ++i) s = fmaf(h2[i], Wo[i * 8 + j], s);
    y[j] = s; mu += s;
  }
  mu *= 0.125f;
  float var = 0.f;
#pragma unroll
  for (int j = 0; j < 8; ++j) { const float d = y[j] - mu; var += d * d; }
  const float rs = rsqrtf(var * 0.125f + 1e-5f);
  const float* g = P + 408; const float* b = P + 400;
#pragma unroll
  for (int j = 0; j < 8; ++j) out[row * 8 + j] = (y[j] - mu) * rs * g[j] + b[j];
}

// segment_sum via f32 global atomics: 8 threads/edge, 4 floats each.
__global__ __launch_bounds__(256) void scatter_add_kernel(
    const float* __restrict__ e, const int* __restrict__ dst,
    float* __restrict__ acc, int ne)
{
  const long t = (long)blockIdx.x * blockDim.x + threadIdx.x;
  if (t >= (long)ne * 8) return;
  const int ed = (int)(t >> 3);
  const int c  = (int)(t & 7) * 4;
  const int d  = dst[ed];
  const float* s = e + (long)ed * 32 + c;
  float* p = acc + (long)d * 32 + c;
  atomicAdd(p + 0, s[0]); atomicAdd(p + 1, s[1]);
  atomicAdd(p + 2, s[2]); atomicAdd(p + 3, s[3]);
}

// ---------------- host side ----------------
struct MP { const float *Wi,*bi,*Wh,*bh,*Wo,*bo,*g,*b; };
// MLP pytree flatten order (sorted keys): hid0.b, hid0.w, in.b, in.w,
// [ln.b, ln.g,] out.b, out.w
static MP mp(const float* P, int base, int fi, int fo, bool norm) {
  MP m;
  m.bh = P + base;          m.Wh = P + base + 64;
  m.bi = P + base + 4160;   m.Wi = P + base + 4224;
  int p = base + 4224 + 64 * fi;
  if (norm) { m.b = P + p; m.g = P + p + fo; p += 2 * fo; }
  else      { m.b = nullptr; m.g = nullptr; }
  m.bo = P + p; m.Wo = P + p + fo;
  return m;
}
#define MLP_ARGS(m) (m).Wi,(m).bi,(m).Wh,(m).bh,(m).Wo,(m).bo,(m).g,(m).b

extern "C" void kernel_launch(void* const* d_in, const int* in_sizes, int n_in,
                              void* d_out, int out_size, void* d_ws, size_t ws_size,
                              hipStream_t stream)
{
  (void)in_sizes; (void)n_in; (void)out_size; (void)ws_size;
  constexpr int NB = 150000, NJ = 40000, N = NB + NJ;
  constexpr int EBB = 400000, EJJ = 100000, EBJ = 100000, EJB = 100000;
  // param base offsets (sorted top-level keys): enc_b2b, enc_b2j, enc_bn,
  // enc_inlet, enc_j2j, enc_jn, enc_outlet, out_b, out_j, proc[0..1]
  constexpr int OFF_ENC_B2B = 0, OFF_ENC_B2J = 6624, OFF_ENC_BN = 13248,
                OFF_ENC_INLET = 20000, OFF_ENC_J2J = 20552, OFF_ENC_JN = 27176,
                OFF_ENC_OUTLET = 33928, OFF_OUT_B = 34480, OFF_OUT_J = 40882,
                OFF_PROC = 47284, PROC_STRIDE = 73024;

  const float* P    = (const float*)d_in[0];
  const float* bx   = (const float*)d_in[1];
  const float* jx   = (const float*)d_in[2];
  const float* inf_ = (const float*)d_in[3];
  const float* outf = (const float*)d_in[4];
  const float* e_bb = (const float*)d_in[5];
  const float* e_jj = (const float*)d_in[6];
  const float* e_bj = (const float*)d_in[7];
  const float* e_jb = (const float*)d_in[8];
  const int* bb_s = (const int*)d_in[9];  const int* bb_d = (const int*)d_in[10];
  const int* jj_s = (const int*)d_in[11]; const int* jj_d = (const int*)d_in[12];
  const int* bj_s = (const int*)d_in[13]; const int* bj_d = (const int*)d_in[14];
  const int* jb_s = (const int*)d_in[15]; const int* jb_d = (const int*)d_in[16];
  float* outp = (float*)d_out;

  float* ws = (float*)d_ws;
  size_t off = 0;
  auto alloc = [&](size_t n) { float* p = ws + off; off += n; return p; };
  float* pb[2]  = { alloc((size_t)NB * 32), alloc((size_t)NB * 32) };
  float* pj[2]  = { alloc((size_t)NJ * 32), alloc((size_t)NJ * 32) };
  float* in_info  = alloc((size_t)N * 8);
  float* out_info = alloc((size_t)N * 8);
  float* ebb[2] = { alloc((size_t)EBB * 32), alloc((size_t)EBB * 32) };
  float* ejj[2] = { alloc((size_t)EJJ * 32), alloc((size_t)EJJ * 32) };
  float* ebj[2] = { alloc((size_t)EBJ * 32), alloc((size_t)EBJ * 32) };
  float* ejb[2] = { alloc((size_t)EJB * 32), alloc((size_t)EJB * 32) };
  float* peb = alloc((size_t)NB * 32);
  float* pej = alloc((size_t)NJ * 32);

  const dim3 blk(256);
  auto g16 = [](int rows) { return dim3((unsigned)((rows + 127) / 128)); };

  // ---- encoders ----
  enc_small_kernel<<<(N + 255) / 256, blk, 0, stream>>>(inf_,  P + OFF_ENC_INLET,  in_info,  N);
  enc_small_kernel<<<(N + 255) / 256, blk, 0, stream>>>(outf,  P + OFF_ENC_OUTLET, out_info, N);
  {
    MP m = mp(P, OFF_ENC_BN, 6, 32, true);
    mlp_wmma<6,32,true,false,0><<<g16(NB), blk, 0, stream>>>(bx,nullptr,nullptr,nullptr,nullptr,nullptr,
        MLP_ARGS(m), nullptr, pb[0], NB);
  }
  {
    MP m = mp(P, OFF_ENC_JN, 6, 32, true);
    mlp_wmma<6,32,true,false,0><<<g16(NJ), blk, 0, stream>>>(jx,nullptr,nullptr,nullptr,nullptr,nullptr,
        MLP_ARGS(m), nullptr, pj[0], NJ);
  }
  {
    MP m = mp(P, OFF_ENC_B2B, 4, 32, true);
    mlp_wmma<4,32,true,false,0><<<g16(EBB), blk, 0, stream>>>(e_bb,nullptr,nullptr,nullptr,nullptr,nullptr,
        MLP_ARGS(m), nullptr, ebb[0], EBB);
  }
  {
    MP m = mp(P, OFF_ENC_J2J, 4, 32, true);
    mlp_wmma<4,32,true,false,0><<<g16(EJJ), blk, 0, stream>>>(e_jj,nullptr,nullptr,nullptr,nullptr,nullptr,
        MLP_ARGS(m), nullptr, ejj[0], EJJ);
  }
  {
    MP m = mp(P, OFF_ENC_B2J, 4, 32, true);   // reference uses enc_b2j for BOTH ebj and ejb
    mlp_wmma<4,32,true,false,0><<<g16(EBJ), blk, 0, stream>>>(e_bj,nullptr,nullptr,nullptr,nullptr,nullptr,
        MLP_ARGS(m), nullptr, ebj[0], EBJ);
    mlp_wmma<4,32,true,false,0><<<g16(EJB), blk, 0, stream>>>(e_jb,nullptr,nullptr,nullptr,nullptr,nullptr,
        MLP_ARGS(m), nullptr, ejb[0], EJB);
  }

  // ---- processor iterations ----
  int cur = 0;
  for (int it = 0; it < 2; ++it) {
    const int nxt = cur ^ 1;
    const int pbase = OFF_PROC + it * PROC_STRIDE;
    // per-iter sorted keys: b2b, b2j, bn, j2b, j2j, jn
    MP m_b2b = mp(P, pbase +     0, 96, 32, true);
    MP m_b2j = mp(P, pbase + 12512, 96, 32, true);
    MP m_bn  = mp(P, pbase + 25024, 80, 32, true);
    MP m_j2b = mp(P, pbase + 36512, 96, 32, true);
    MP m_j2j = mp(P, pbase + 49024, 96, 32, true);
    MP m_jn  = mp(P, pbase + 61536, 80, 32, true);

    mlp_wmma<96,32,true,true,1><<<g16(EBB), blk, 0, stream>>>(ebb[cur], pb[cur], pb[cur], nullptr,
        bb_s, bb_d, MLP_ARGS(m_b2b), ebb[cur], ebb[nxt], EBB);
    mlp_wmma<96,32,true,true,1><<<g16(EJJ), blk, 0, stream>>>(ejj[cur], pj[cur], pj[cur], nullptr,
        jj_s, jj_d, MLP_ARGS(m_j2j), ejj[cur], ejj[nxt], EJJ);
    mlp_wmma<96,32,true,true,1><<<g16(EBJ), blk, 0, stream>>>(ebj[cur], pb[cur], pj[cur], nullptr,
        bj_s, bj_d, MLP_ARGS(m_b2j), ebj[cur], ebj[nxt], EBJ);
    mlp_wmma<96,32,true,true,1><<<g16(EJB), blk, 0, stream>>>(ejb[cur], pj[cur], pb[cur], nullptr,
        jb_s, jb_d, MLP_ARGS(m_j2b), ejb[cur], ejb[nxt], EJB);

    hipMemsetAsync(peb, 0, (size_t)NB * 32 * sizeof(float), stream);
    hipMemsetAsync(pej, 0, (size_t)NJ * 32 * sizeof(float), stream);
    scatter_add_kernel<<<(unsigned)(((long)EBB * 8 + 255) / 256), blk, 0, stream>>>(ebb[nxt], bb_d, peb, EBB);
    scatter_add_kernel<<<(unsigned)(((long)EJB * 8 + 255) / 256), blk, 0, stream>>>(ejb[nxt], jb_d, peb, EJB);
    scatter_add_kernel<<<(unsigned)(((long)EJJ * 8 + 255) / 256), blk, 0, stream>>>(ejj[nxt], jj_d, pej, EJJ);
    scatter_add_kernel<<<(unsigned)(((long)EBJ * 8 + 255) / 256), blk, 0, stream>>>(ebj[nxt], bj_d, pej, EBJ);

    mlp_wmma<80,32,true,true,2><<<g16(NB), blk, 0, stream>>>(pb[cur], peb,
        in_info, out_info, nullptr, nullptr, MLP_ARGS(m_bn), pb[cur], pb[nxt], NB);
    mlp_wmma<80,32,true,true,2><<<g16(NJ), blk, 0, stream>>>(pj[cur], pej,
        in_info + (size_t)NB * 8, out_info + (size_t)NB * 8, nullptr, nullptr,
        MLP_ARGS(m_jn), pj[cur], pj[nxt], NJ);
    cur = nxt;
  }

  // ---- decoders -> d_out ----
  {
    MP m = mp(P, OFF_OUT_B, 32, 2, false);
    mlp_wmma<32,2,false,false,0><<<g16(NB), blk, 0, stream>>>(pb[cur],nullptr,nullptr,nullptr,nullptr,nullptr,
        MLP_ARGS(m), nullptr, outp, NB);
  }
  {
    MP m = mp(P, OFF_OUT_J, 32, 2, false);
    mlp_wmma<32,2,false,false,0><<<g16(NJ), blk, 0, stream>>>(pj[cur],nullptr,nullptr,nullptr,nullptr,nullptr,
        MLP_ARGS(m), nullptr, outp + (size_t)NB * 2, NJ);
  }
}